// SimpleFSWEncoder_22780506538153
// MI455X (gfx1250) — compile-verified
//
#include <hip/hip_runtime.h>
#include <math.h>

typedef float v2f __attribute__((ext_vector_type(2)));
typedef float v8f __attribute__((ext_vector_type(8)));

#define D_   512
#define D2_  1024
#define KC_  32      // K-chunk staged in LDS per iteration (= 16 k-pairs)

__device__ __forceinline__ float gelu_exact(float x) {
    // exact GELU: 0.5 * x * (1 + erf(x / sqrt(2)))
    return 0.5f * x * (1.0f + erff(x * 0.7071067811865475f));
}

// ---------------------------------------------------------------------------
// Kernel 0: interleave weights into K-pair layout:
//   Wi[(k>>1)*1024 + c*2 + (k&1)] = W[k*512 + c]
// so a WMMA B fragment (B[2p][c], B[2p+1][c]) is 8 contiguous bytes.
// ---------------------------------------------------------------------------
__global__ void fsw_interleave_kernel(const float* __restrict__ W,
                                      float* __restrict__ Wi, int K) {
    const int idx = blockIdx.x * 256 + threadIdx.x;   // over (K/2)*512 pairs
    const int p = idx >> 9;          // k-pair index
    const int c = idx & 511;         // column
    if (p < (K >> 1)) {
        float2 v;
        v.x = W[(size_t)(2 * p) * D_ + c];
        v.y = W[(size_t)(2 * p + 1) * D_ + c];
        *(float2*)(Wi + (size_t)p * (2 * D_) + c * 2) = v;
    }
}

// ---------------------------------------------------------------------------
// Kernel 1: ragged mean-pool. One block (128 threads) per segment.
// pooled[s, 0:512]    = mean over segment tokens of embed_table[symbol_ids[t]]
// pooled[s, 512:1024] = mean(p0)*W_pos[0] + mean(p1)*W_pos[1] + b_pos
// ---------------------------------------------------------------------------
__global__ void fsw_pool_kernel(const int* __restrict__ symbol_ids,
                                const float* __restrict__ positions,
                                const int* __restrict__ segment_ids,
                                const float* __restrict__ embed_table,
                                const float* __restrict__ W_pos,   // [2,512]
                                const float* __restrict__ b_pos,   // [512]
                                float* __restrict__ pooled,        // [B,1024]
                                int T) {
    __shared__ int  seg_range[2];
    __shared__ float red0[128];
    __shared__ float red1[128];

    const int s   = blockIdx.x;
    const int tid = threadIdx.x;

    if (tid == 0) {
        // lower_bound(segment_ids, s) and lower_bound(segment_ids, s+1)
        int lo = 0, hi = T;
        while (lo < hi) { int mid = (lo + hi) >> 1; if (segment_ids[mid] < s) lo = mid + 1; else hi = mid; }
        seg_range[0] = lo;
        hi = T;
        while (lo < hi) { int mid = (lo + hi) >> 1; if (segment_ids[mid] < s + 1) lo = mid + 1; else hi = mid; }
        seg_range[1] = lo;
    }
    __syncthreads();

    const int start = seg_range[0];
    const int end   = seg_range[1];
    const int count = end - start;
    const int d     = tid * 4;                 // this thread's 4 dims in each half
    float* out_row  = pooled + (size_t)s * D2_;

    if (count <= 0) {
        #pragma unroll
        for (int i = 0; i < 4; ++i) {
            out_row[d + i]       = 0.0f;
            out_row[D_ + d + i]  = 0.0f;
        }
        return;
    }

    // embedding half: accumulate 4 dims per thread over tokens (L2-resident table)
    float4 acc = make_float4(0.f, 0.f, 0.f, 0.f);
    for (int t = start; t < end; ++t) {
        const int sid = symbol_ids[t];
        const float4 e = *(const float4*)(embed_table + (size_t)sid * D_ + d);
        acc.x += e.x; acc.y += e.y; acc.z += e.z; acc.w += e.w;
    }

    // position sums (block reduce)
    float p0 = 0.f, p1 = 0.f;
    for (int t = start + tid; t < end; t += 128) {
        p0 += positions[2 * t];
        p1 += positions[2 * t + 1];
    }
    red0[tid] = p0; red1[tid] = p1;
    __syncthreads();
    for (int off = 64; off > 0; off >>= 1) {
        if (tid < off) { red0[tid] += red0[tid + off]; red1[tid] += red1[tid + off]; }
        __syncthreads();
    }
    const float inv = 1.0f / (float)count;
    const float mp0 = red0[0] * inv;
    const float mp1 = red1[0] * inv;

    #pragma unroll
    for (int i = 0; i < 4; ++i) {
        out_row[d + i]      = (&acc.x)[i] * inv;
        out_row[D_ + d + i] = mp0 * W_pos[d + i] + mp1 * W_pos[D_ + d + i] + b_pos[d + i];
    }
}

// ---------------------------------------------------------------------------
// Kernels 2/3: C = act(A[MxK] * B[Kx512] + bias), act = exact GELU / identity.
// B is pre-interleaved (Bi, K-pair layout). Wave32 WMMA f32 16x16x4.
// Block = 4 waves, 64(M) x 64(N) C tile; each wave owns a 16x64 strip.
//
// Pipelining (per KC_=32 chunk, all double-buffered):
//   - Bi slice [16 pairs x 64 cols x 2] staged to LDS via
//     GLOBAL_LOAD_ASYNC_TO_LDS_B128 (ASYNCcnt; in-order done =>
//     s_wait_asynccnt 4 waits only the previous chunk)
//   - A fragments for the *next* chunk preloaded into registers (8x b64
//     clause) before the current chunk's 32 WMMAs
//   - B fragment = one aligned ds_load_b64 (pair is contiguous in LDS),
//     no VGPR repacking.
//
// ISA §7.12.2 layouts (wave32):
//   A 16x4 : lane m=lane&15; v0 = A[m][2*kh], v1 = A[m][2*kh+1], kh=lane>>4
//   B 4x16 : lane n=lane&15; v0 = B[2*kh][n], v1 = B[2*kh+1][n]
//   C 16x16: vgpr r -> row = r + 8*kh, col = lane&15
// ---------------------------------------------------------------------------
template <int K, bool GELU>
__global__ void fsw_gemm_kernel(const float* __restrict__ A,     // [M x K]
                                const float* __restrict__ Bi,    // interleaved [K/2 x 512 x 2]
                                const float* __restrict__ bias,  // [512]
                                float* __restrict__ C) {         // [M x 512]
    __shared__ float smem[2][KC_ * 64];       // 2 x 8KB double buffer

    const int tid   = threadIdx.x;            // 0..127
    const int lane  = tid & 31;
    const int wave  = tid >> 5;
    const int mtile = blockIdx.y * 64 + wave * 16;
    const int ntile = blockIdx.x * 64;
    const int mn    = lane & 15;   // row within A tile / col within B sub-tile
    const int kh    = lane >> 4;   // which K-pair this half-wave holds

    v8f acc[4] = {};
    v2f a0[KC_ / 4], a1[KC_ / 4];             // per-chunk A fragment buffers

    const float* arow = A + (size_t)(mtile + mn) * K + 2 * kh;

    // async-stage one chunk (16 pair-rows x 128 floats) of Bi into smem[buf]
    auto stage_chunk = [&](int k0, int buf) {
        const float* base = Bi + (size_t)(k0 >> 1) * D2_ + ntile * 2;
        #pragma unroll
        for (int r = 0; r < 4; ++r) {
            const int i4 = tid + r * 128;          // 0..511 float4 slots
            const int p  = i4 >> 5;                // 0..15 k-pair row in chunk
            const int w  = (i4 & 31) * 4;          // 0..124 float offset in row
            const float* g = base + (size_t)p * D2_ + w;
            const unsigned lds = (unsigned)(uintptr_t)&smem[buf][p * 128 + w];
            asm volatile("global_load_async_to_lds_b128 %0, %1, off"
                         :: "v"(lds), "v"((unsigned long long)(uintptr_t)g)
                         : "memory");
        }
    };

    // preload this chunk's A fragments into registers (8 x global b64 clause)
    auto load_a = [&](int k0, v2f (&ar)[KC_ / 4]) {
        #pragma unroll
        for (int q = 0; q < KC_ / 4; ++q)
            ar[q] = *(const v2f*)(arow + k0 + q * 4);
    };

    // 32 WMMAs of one chunk; B fragments are single ds_load_b64 from LDS
    auto compute = [&](const v2f (&ar)[KC_ / 4], const float* sb) {
        #pragma unroll
        for (int kq = 0; kq < KC_; kq += 4) {
            #pragma unroll
            for (int j = 0; j < 4; ++j) {
                const int bofs = ((kq >> 1) + kh) * 128 + (j * 16 + mn) * 2;
                v2f b = *(const v2f*)(sb + bofs);
                acc[j] = __builtin_amdgcn_wmma_f32_16x16x4_f32(
                    /*neg_a=*/false, ar[kq / 4], /*neg_b=*/false, b,
                    /*c_mod=*/(short)0, acc[j],
                    /*reuse_a=*/false, /*reuse_b=*/false);
            }
        }
    };

    constexpr int NCHUNK = K / KC_;           // 32 or 16 (even)

    stage_chunk(0, 0);
    load_a(0, a0);

    for (int i = 0; i < NCHUNK; i += 2) {
        const int k0 = i * KC_;

        // ---- even chunk i: buf 0 / a0; prefetch chunk i+1 into buf 1 / a1 ----
        stage_chunk(k0 + KC_, 1);
        load_a(k0 + KC_, a1);
        __builtin_prefetch(arow + k0 + 2 * KC_, 0, 1);   // global_prefetch_b8
        asm volatile("s_wait_asynccnt 4" ::: "memory");  // chunk i done (in-order)
        __syncthreads();
        compute(a0, &smem[0][0]);
        __syncthreads();

        // ---- odd chunk i+1: buf 1 / a1; prefetch chunk i+2 into buf 0 / a0 ----
        if (i + 2 < NCHUNK) {
            stage_chunk(k0 + 2 * KC_, 0);
            load_a(k0 + 2 * KC_, a0);
            asm volatile("s_wait_asynccnt 4" ::: "memory");
        } else {
            asm volatile("s_wait_asynccnt 0" ::: "memory");
        }
        __syncthreads();
        compute(a1, &smem[1][0]);
        __syncthreads();
    }

    #pragma unroll
    for (int j = 0; j < 4; ++j) {
        const int col = ntile + j * 16 + mn;
        const float bv = bias[col];
        #pragma unroll
        for (int r = 0; r < 8; ++r) {
            const int row = mtile + r + 8 * kh;
            float v = acc[j][r] + bv;
            if (GELU) v = gelu_exact(v);
            C[(size_t)row * D_ + col] = v;
        }
    }
}

extern "C" void kernel_launch(void* const* d_in, const int* in_sizes, int n_in,
                              void* d_out, int out_size, void* d_ws, size_t ws_size,
                              hipStream_t stream) {
    const int*   symbol_ids  = (const int*)d_in[0];
    const float* positions   = (const float*)d_in[1];
    const int*   segment_ids = (const int*)d_in[2];
    // d_in[3] = batch_size scalar on device; derive B from out_size instead
    const float* embed_table = (const float*)d_in[4];
    const float* W_pos       = (const float*)d_in[5];
    const float* b_pos       = (const float*)d_in[6];
    const float* W1          = (const float*)d_in[7];
    const float* b1          = (const float*)d_in[8];
    const float* W2          = (const float*)d_in[9];
    const float* b2          = (const float*)d_in[10];
    float*       out         = (float*)d_out;

    const int T  = in_sizes[0];
    const int Bn = out_size / D_;              // 8192

    float* pooled = (float*)d_ws;              // [Bn, 1024]       32 MB
    float* h      = pooled + (size_t)Bn * D2_; // [Bn, 512]        16 MB
    float* W1i    = h      + (size_t)Bn * D_;  // interleaved W1    2 MB
    float* W2i    = W1i    + (size_t)D2_ * D_; // interleaved W2    1 MB

    fsw_interleave_kernel<<<(D2_ / 2 * D_) / 256, 256, 0, stream>>>(W1, W1i, D2_);
    fsw_interleave_kernel<<<(D_  / 2 * D_) / 256, 256, 0, stream>>>(W2, W2i, D_);

    fsw_pool_kernel<<<Bn, 128, 0, stream>>>(symbol_ids, positions, segment_ids,
                                            embed_table, W_pos, b_pos, pooled, T);

    dim3 grid(D_ / 64, Bn / 64);               // (8, 128)
    fsw_gemm_kernel<D2_, true ><<<grid, 128, 0, stream>>>(pooled, W1i, b1, h);
    fsw_gemm_kernel<D_,  false><<<grid, 128, 0, stream>>>(h,      W2i, b2, out);
}